// LowpassFilter_19301583028616
// MI455X (gfx1250) — compile-verified
//
#include <hip/hip_runtime.h>
#include <hip/hip_bf16.h>

typedef __attribute__((ext_vector_type(2))) float v2f;
typedef __attribute__((ext_vector_type(8))) float v8f;

// ---- filter constants (match reference: np.float32(c / a0) from float64 math) ----
// w0 = 2*pi*4000/32000 = pi/4 ; sin(w0)=cos(w0)=sqrt(2)/2
#define SINW0 0.70710678118654752440
#define COSW0 0.70710678118654752440
#define ALPHA_ (SINW0 / (2.0 * 0.707))
#define A0D    (1.0 + ALPHA_)

#define KB0 ((float)(((1.0 - COSW0) / 2.0) / A0D))
#define KB1 ((float)((1.0 - COSW0) / A0D))
#define KB2 ((float)(((1.0 - COSW0) / 2.0) / A0D))
#define KA1 ((float)((-2.0 * COSW0) / A0D))
#define KA2 ((float)((1.0 - ALPHA_) / A0D))

#define NTAPS 64     // |pole|~0.577 -> tail beyond 64 taps ~5e-16, below fp32 eps
#define TBATCH 256
#define TLEN   160000
#define ROWS_PER_BLK 16
#define TIME_PER_BLK 128           // 8 waves * 16 time steps
#define WIN (TIME_PER_BLK + NTAPS) // 192 input columns staged in LDS
#define LDS_STRIDE 193             // odd stride -> low-conflict across 64 banks
#define NCHUNK 20                  // 80-wide reduction window / K=4 per WMMA
#define BTAB_FLOATS (NCHUNK * 32 * 2)  // 1280 floats: per-(chunk,lane) B pairs

// ---------------------------------------------------------------------------
// Kernel 1 (1 block): build the per-lane B-operand table for the banded
// Toeplitz tap matrix B[u,n] = h[n + 64 - u] (zero outside [0,64)).
// Layout: ws[(p*32 + lane)*2 + comp], comp selects K = 4p+2*half (+comp).
// ---------------------------------------------------------------------------
__global__ __launch_bounds__(64)
void LowpassFilter_build_taps(float* __restrict__ ws) {
    __shared__ float sh[NTAPS];
    if (threadIdx.x == 0) {
        float x1 = 0.f, x2 = 0.f, y1 = 0.f, y2 = 0.f;
        for (int n = 0; n < NTAPS; ++n) {
            float xn = (n == 0) ? 1.0f : 0.0f;
            float yn = KB0 * xn + KB1 * x1 + KB2 * x2 - KA1 * y1 - KA2 * y2;
            sh[n] = yn;
            x2 = x1; x1 = xn; y2 = y1; y1 = yn;
        }
    }
    __syncthreads();
    for (int i = threadIdx.x; i < BTAB_FLOATS; i += 64) {
        int p    = i >> 6;          // chunk
        int rem  = i & 63;
        int lane = rem >> 1;
        int comp = rem & 1;
        int n    = lane & 15;       // B column (time within sub-tile)
        int half = lane >> 4;       // lane half -> K pair select
        int u    = 4 * p + 2 * half + comp;
        int j    = n + NTAPS - u;
        ws[i] = ((unsigned)j < (unsigned)NTAPS) ? sh[j] : 0.0f;
    }
}

// ---------------------------------------------------------------------------
// Kernel 2: FIR-as-GEMM. Each block: 16 batch rows x 128 time steps.
// Each wave: one 16x16 output tile via 20 chained V_WMMA_F32_16X16X4_F32.
// ---------------------------------------------------------------------------
__global__ __launch_bounds__(256)
void LowpassFilter_fir_wmma_kernel(const float* __restrict__ x,
                                   float* __restrict__ y,
                                   const float* __restrict__ btab) {
    __shared__ float sx[ROWS_PER_BLK * LDS_STRIDE]; // input slab [16][193]
    __shared__ v2f   sB[NCHUNK * 32];               // per-(chunk,lane) B pairs

    const int tid = threadIdx.x;
    const int b0r = blockIdx.y * ROWS_PER_BLK;      // first batch row of tile
    const int t0  = blockIdx.x * TIME_PER_BLK;      // first output time of tile

    // --- stage B table (L2-resident, 5KB) into LDS ---
    {
        const v2f* bt = (const v2f*)btab;
        for (int i = tid; i < NCHUNK * 32; i += 256) sB[i] = bt[i];
    }

    // --- cooperative global -> LDS load of x[b0r:+16, t0-64 : t0+128) ---
    // zero-pad t<0 (matches reference's zero initial conditions)
    for (int i = tid; i < ROWS_PER_BLK * WIN; i += 256) {
        int r = i / WIN;
        int c = i - r * WIN;
        int g = t0 - NTAPS + c;
        float v = 0.0f;
        if (g >= 0) v = x[(size_t)(b0r + r) * TLEN + g];
        sx[r * LDS_STRIDE + c] = v;
    }
    __syncthreads();

    // --- per-wave 16x16 sub-tile ---
    const int lane = tid & 31;
    const int wave = tid >> 5;          // 0..7 -> sub-tile time offset
    const int tw   = wave * 16;
    const int m    = lane & 15;         // A row (batch) / D column (time)
    const int half = lane >> 4;         // lane half selects K pair / M+8

    v8f acc = {0.f, 0.f, 0.f, 0.f, 0.f, 0.f, 0.f, 0.f};

    const float* rowA = &sx[m * LDS_STRIDE + tw + 2 * half];
    const v2f*   laneB = &sB[lane];

#pragma unroll
    for (int p = 0; p < NCHUNK; ++p) {
        // A[m, u] = x[row m, t0 + tw - 64 + u], u = 4p + 2*half (+1)
        v2f a;
        a.x = rowA[4 * p];
        a.y = rowA[4 * p + 1];

        // B pair for this (chunk, lane): single aligned b64 LDS load
        v2f b = laneB[p * 32];

        acc = __builtin_amdgcn_wmma_f32_16x16x4_f32(
            /*neg_a=*/false, a, /*neg_b=*/false, b,
            /*c_mod=*/(short)0, acc, /*reuse_a=*/false, /*reuse_b=*/false);
    }

    // --- store D: lane (l&15)=N(time), VGPR v = M row (lanes>=16 -> M+8) ---
    const int colg = t0 + tw + m;
#pragma unroll
    for (int v = 0; v < 8; ++v) {
        int rowg = b0r + v + 8 * half;
        y[(size_t)rowg * TLEN + colg] = acc[v];
    }
}

extern "C" void kernel_launch(void* const* d_in, const int* in_sizes, int n_in,
                              void* d_out, int out_size, void* d_ws, size_t ws_size,
                              hipStream_t stream) {
    const float* clip = (const float*)d_in[0];
    float* out = (float*)d_out;
    float* btab = (float*)d_ws;   // 1280 floats (5KB) of B-operand table

    LowpassFilter_build_taps<<<1, 64, 0, stream>>>(btab);

    dim3 grid(TLEN / TIME_PER_BLK, TBATCH / ROWS_PER_BLK, 1); // 1250 x 16
    dim3 block(256, 1, 1);
    LowpassFilter_fir_wmma_kernel<<<grid, block, 0, stream>>>(clip, out, btab);
}